// Set2SetLoss_4578435137647
// MI455X (gfx1250) — compile-verified
//
#include <hip/hip_runtime.h>

typedef __attribute__((ext_vector_type(2))) float v2f;
typedef __attribute__((ext_vector_type(8))) float v8f;

#define B_    128
#define N_    192
#define D_    3
#define TILE  16
#define TPB   12                      // 192/16 tiles per dimension
#define TILES_PER_BATCH (TPB * TPB)   // 144
#define INFF  __builtin_huge_valf()

// ---------------------------------------------------------------------------
// Fused kernel: one block per batch (128 blocks x 256 threads = 8 wave32).
//  Phase 1: 8 waves build the 192x192 cost matrix in LDS via
//           V_WMMA_F32_16X16X4_F32 (||x||^2 + ||y||^2 - 2 y.x, K padded 3->4).
//  Phase 2: Jonker-Volgenant Hungarian, LDS-resident cost, register-resident
//           per-column state (v, minv, used), wave32 shfl_xor argmin.
//           Only 2 __syncthreads per augmenting step.
//  Phase 3: gather matched pairs, per-batch partial sums (LDS tree, reusing
//           the cost LDS as scratch).
// LDS: 144 KB dynamic (cost) + ~2.5 KB static -> one workgroup per WGP.
// ---------------------------------------------------------------------------
__global__ __launch_bounds__(256) void set2set_fused_kernel(
    const float* __restrict__ inp,   // (B, N, 3)  columns p
    const float* __restrict__ tgt,   // (B, N, 3)  rows t
    const float* __restrict__ mask,  // (B, N, 2)
    float* __restrict__ partial)     // (B, 4)
{
  extern __shared__ float cost[];    // N_*N_ floats = 147456 B

  __shared__ float u[N_ + 1];        // row duals (cross-thread indexed)
  __shared__ int   p[N_ + 1];        // column -> matched row
  __shared__ short way[N_ + 1];      // alternating-path predecessors
  __shared__ float wval[8];          // per-wave argmin value
  __shared__ int   widx[8];          // per-wave argmin index

  const int b    = blockIdx.x;
  const int tid  = threadIdx.x;
  const int wv   = tid >> 5;         // wave 0..7
  const int lane = tid & 31;
  const float* X = inp + (size_t)b * N_ * D_;
  const float* Y = tgt + (size_t)b * N_ * D_;

  // warm L2/L0 for this batch's point data
  if (tid < 18) {
    __builtin_prefetch(X + tid * 32, 0, 3);
    __builtin_prefetch(Y + tid * 32, 0, 3);
  }

  // ---------------- Phase 1: WMMA cost tiles -> LDS -----------------------
  {
    const int half = lane >> 4;      // 0: K=0,1   1: K=2,3
    const int l    = lane & 15;
    const int k0   = 2 * half;       // 0 or 2 (always < 3)
    const int k1   = k0 + 1;         // 1 or 3 (3 = zero pad)
    for (int tl = wv; tl < TILES_PER_BATCH; tl += 8) {   // uniform trip count
      int trow = (tl / TPB) * TILE;
      int pcol = (tl % TPB) * TILE;

      v2f a;                         // A (16x4): row M=l, K pair (k0,k1)
      a.x = Y[(trow + l) * D_ + k0];
      a.y = (k1 < D_) ? Y[(trow + l) * D_ + k1] : 0.0f;
      v2f bm;                        // B (4x16): col N=l, same K split
      bm.x = X[(pcol + l) * D_ + k0];
      bm.y = (k1 < D_) ? X[(pcol + l) * D_ + k1] : 0.0f;

      v8f c = {};
      v8f d = __builtin_amdgcn_wmma_f32_16x16x4_f32(
          false, a, false, bm, (short)0, c, false, false);

      float x0 = X[(pcol + l) * D_ + 0], x1 = X[(pcol + l) * D_ + 1], x2 = X[(pcol + l) * D_ + 2];
      float nx = x0 * x0 + x1 * x1 + x2 * x2;
      float y0 = Y[(trow + l) * D_ + 0], y1 = Y[(trow + l) * D_ + 1], y2 = Y[(trow + l) * D_ + 2];
      float ny_own = y0 * y0 + y1 * y1 + y2 * y2;
      float mcol = mask[((size_t)b * N_ + (pcol + l)) * 2 + 1];

#pragma unroll
      for (int r = 0; r < 8; ++r) {
        int M = r + 8 * half;                       // C/D row in VGPR r
        float nyM = __shfl(ny_own, M, 32);
        cost[(trow + M) * N_ + pcol + l] =
            (nx + nyM - 2.0f * d[r]) * (1.0f / 3.0f) * mcol;
      }
    }
  }

  if (tid <= N_) { u[tid] = 0.0f; p[tid] = 0; }
  float v_r = 0.0f;                  // column dual (thread tid owns column tid)
  __syncthreads();

  // ---------------- Phase 2: Hungarian (2 barriers / step) ----------------
  const bool isCol = (tid >= 1 && tid <= N_);
  for (int i = 1; i <= N_; ++i) {
    if (tid == 0) p[0] = i;
    float minv_r = INFF;
    bool  used_r = false;
    __syncthreads();                 // p[0] visible; prior-row writes settled

    int j0 = 0;                      // uniform across threads
    while (true) {
      if (tid == j0) used_r = true;  // only thread j0 ever reads its own flag
      int   i0  = p[j0];             // broadcast LDS read (stable since barrier)
      float ui0 = u[i0];

      // phase A: relax own column, stage (val, idx) for argmin
      float myv = INFF;
      if (isCol && !used_r) {
        float cur = cost[(i0 - 1) * N_ + (tid - 1)] - ui0 - v_r;
        if (cur < minv_r) { minv_r = cur; way[tid] = (short)j0; }
        myv = minv_r;
      }
      int myj = tid;

      // wave32 lexicographic-min butterfly (barrier-free)
#pragma unroll
      for (int off = 16; off > 0; off >>= 1) {
        float ov = __shfl_xor(myv, off, 32);
        int   oj = __shfl_xor(myj, off, 32);
        if (ov < myv || (ov == myv && oj < myj)) { myv = ov; myj = oj; }
      }
      if (lane == 0) { wval[wv] = myv; widx[wv] = myj; }
      __syncthreads();               // barrier 1: per-wave minima visible

      // cross-wave min, computed redundantly by every thread (broadcast reads)
      float delta = wval[0];
      int   j1    = widx[0];
#pragma unroll
      for (int w = 1; w < 8; ++w) {
        float ov = wval[w]; int oj = widx[w];
        if (ov < delta || (ov == delta && oj < j1)) { delta = ov; j1 = oj; }
      }
      bool done = (p[j1] == 0);      // p untouched until augmentation

      // phase B: dual updates (u[p[tid]] distinct over used cols -> race-free)
      if (tid <= N_) {
        if (used_r)      { u[p[tid]] += delta; v_r -= delta; }
        else if (tid >= 1) minv_r -= delta;
      }
      j0 = j1;
      __syncthreads();               // barrier 2: u updates visible next step
      if (done) break;
    }

    // augment along the alternating path (serial, short)
    if (tid == 0) {
      int ja = j0;
      while (ja != 0) { int jb = way[ja]; p[ja] = p[jb]; ja = jb; }
    }
    __syncthreads();
  }

  // ---------------- Phase 3: gather + per-batch reduction -----------------
  // Column j (1..N) matched to target row t = p[j]-1, input c = j-1.
  float* s0 = cost;
  float* s1 = cost + N_;
  float* s2 = cost + 2 * N_;
  float* s3 = cost + 3 * N_;
  if (isCol) {
    int t = p[tid] - 1;
    int c = tid - 1;
    const float* x = X + c * D_;
    const float* y = Y + t * D_;
    float m = mask[((size_t)b * N_ + c) * 2 + 1];
    float d0 = x[0] - y[0]; d0 *= d0;
    float d1 = x[1] - y[1]; d1 *= d1;
    float d2 = x[2] - y[2]; d2 *= d2;
    s0[t] = (d0 + d1 + d2) * (1.0f / 3.0f) * m;
    s1[t] = d0 * m;
    s2[t] = d1 * m;
    s3[t] = d2 * m;
  }
  __syncthreads();
  for (int s = 96; s >= 3; s >>= 1) {   // 96,48,24,12,6,3
    if (tid < s) { s0[tid] += s0[tid+s]; s1[tid] += s1[tid+s]; s2[tid] += s2[tid+s]; s3[tid] += s3[tid+s]; }
    __syncthreads();
  }
  if (tid == 0) {
    partial[(size_t)b * 4 + 0] = s0[0] + s0[1] + s0[2];
    partial[(size_t)b * 4 + 1] = s1[0] + s1[1] + s1[2];
    partial[(size_t)b * 4 + 2] = s2[0] + s2[1] + s2[2];
    partial[(size_t)b * 4 + 3] = s3[0] + s3[1] + s3[2];
  }
}

// ---------------------------------------------------------------------------
// Final reduction over batches, scale by 1/(B*T)
// ---------------------------------------------------------------------------
__global__ __launch_bounds__(128) void final_kernel(
    const float* __restrict__ partial, float* __restrict__ out)
{
  int t = threadIdx.x;
  __shared__ float s0[128], s1[128], s2[128], s3[128];
  s0[t] = partial[(size_t)t * 4 + 0];
  s1[t] = partial[(size_t)t * 4 + 1];
  s2[t] = partial[(size_t)t * 4 + 2];
  s3[t] = partial[(size_t)t * 4 + 3];
  __syncthreads();
  for (int s = 64; s > 0; s >>= 1) {
    if (t < s) { s0[t] += s0[t+s]; s1[t] += s1[t+s]; s2[t] += s2[t+s]; s3[t] += s3[t+s]; }
    __syncthreads();
  }
  if (t == 0) {
    const float scale = 1.0f / ((float)B_ * (float)N_);
    out[0] = s0[0] * scale;
    out[1] = s1[0] * scale;
    out[2] = s2[0] * scale;
    out[3] = s3[0] * scale;
  }
}

// ---------------------------------------------------------------------------
extern "C" void kernel_launch(void* const* d_in, const int* in_sizes, int n_in,
                              void* d_out, int out_size, void* d_ws, size_t ws_size,
                              hipStream_t stream) {
  (void)in_sizes; (void)n_in; (void)out_size; (void)ws_size;
  const float* inp  = (const float*)d_in[0];   // (128,192,3)
  const float* tgt  = (const float*)d_in[1];   // (128,192,3)
  const float* mask = (const float*)d_in[2];   // (128,192,2)
  float* out     = (float*)d_out;              // 4 floats
  float* partial = (float*)d_ws;               // B*4 f32

  const size_t dynLds = (size_t)N_ * N_ * sizeof(float);   // 147456 B
  set2set_fused_kernel<<<B_, 256, dynLds, stream>>>(inp, tgt, mask, partial);
  final_kernel<<<1, 128, 0, stream>>>(partial, out);
}